// DTWLAYER_69080253989228
// MI455X (gfx1250) — compile-verified
//
#include <hip/hip_runtime.h>

// Soft-DTW layer for MI455X (gfx1250, wave32).
//   x:      [B=64, T=512] f32
//   protos: [F=32, K=64]  f32
//   out:    [B, F, 1]     f32  (flat b*F + f)
//
// One wave (32 lanes) per (b,f) problem; anti-diagonal wavefront DP,
// 2 columns per lane. Cost tiles log2e*(x - p)^2 computed in f32 via
// V_WMMA_F32_16X16X4_F32 using the identity
//   (x - p)^2 * L2E = [x^2, x, 1, 0] . [L2E, -2p*L2E, p^2*L2E, 0]^T
// The entire DP runs in the base-2 domain (D' = D * log2e), so softmin is
//   m - log2(2^(m-a) + 2^(m-b) + 2^(m-c))
// with bare v_exp_f32 / v_log_f32 and no per-exp base conversion; the single
// conversion back (* ln2) is folded into the final output scale.
// One WMMA = one 16x16 cost tile, streamed through an 80-row LDS ring buffer;
// all ring addressing is incremental pointer arithmetic (no modulo, no shifts
// in the hot loop).

typedef __attribute__((ext_vector_type(2))) float v2f;
typedef __attribute__((ext_vector_type(8))) float v8f;

#define BIGV   1.0e9f
#define T_LEN  512
#define K_LEN  64
#define F_NUM  32
#define B_NUM  64
#define RING   80                    // rows live in [d-63, d+15] -> 79; 80 is safe
#define RINGSZ (RING * K_LEN)        // 5120 floats
#define N_DIAG (T_LEN + K_LEN - 1)   // 575
#define L2E    1.44269504088896340736f
#define LN2    0.69314718055994530942f

__device__ __forceinline__ float softmin3_b2(float a, float b, float c) {
    // Base-2 soft-min: operands are already scaled by log2e. exp2 args <= 0,
    // log2 arg in [1, 3] -> raw trans ops are safe (no fixups needed).
    float m = fminf(a, fminf(b, c));
    float s = __builtin_amdgcn_exp2f(m - a)
            + __builtin_amdgcn_exp2f(m - b)
            + __builtin_amdgcn_exp2f(m - c);
    return m - __builtin_amdgcn_logf(s);
}

__global__ __launch_bounds__(32)
void dtw_softdtw_kernel(const float* __restrict__ x,
                        const float* __restrict__ protos,
                        float* __restrict__ out) {
    __shared__ float ldsC[RINGSZ];         // 20 KB cost ring buffer

    const int bf   = blockIdx.x;           // 0 .. B*F-1
    const int b    = bf / F_NUM;
    const int f    = bf % F_NUM;
    const int lane = threadIdx.x;          // 0..31 (wave32)
    const int lc   = lane & 15;

    // Prefetch this wave's x row (global_prefetch_b8 on gfx1250).
    __builtin_prefetch(&x[b * T_LEN], 0, 3);

    // B operands for the 4 k-tiles (4x16 f32, mirrors ISA 16x4 A layout):
    //   VGPR0: lanes 0-15 -> K=0 row, lanes 16-31 -> K=2 row
    //   VGPR1: lanes 0-15 -> K=1 row, lanes 16-31 -> K=3 row
    // B[:,n] = [L2E, -2*p_n*L2E, p_n^2*L2E, 0]  (base-2 cost scaling folded in)
    v2f bvec[4];
#pragma unroll
    for (int t = 0; t < 4; ++t) {
        float pv = protos[f * K_LEN + t * 16 + lc];
        bvec[t].x = (lane < 16) ? L2E : pv * pv * L2E;
        bvec[t].y = (lane < 16) ? ((-2.0f * L2E) * pv) : 0.0f;
    }

    // This lane owns columns j0 = 2*lane, j1 = 2*lane+1.
    const int j0 = 2 * lane;
    const int j1 = 2 * lane + 1;

    // Sakoe-Chiba band as integer row bounds per column (bit-exact vs the
    // reference's f32 mask: |63i - 511j| / 32193 is never within f32 rounding
    // distance of 0.2, so no boundary ambiguity). Evaluated once in f64.
    const double bw = 0.2;
    int ilo0 = (int)ceil(((double)j0 / 63.0 - bw) * 511.0); if (ilo0 < 0) ilo0 = 0;
    int ihi0 = (int)floor(((double)j0 / 63.0 + bw) * 511.0); if (ihi0 > T_LEN - 1) ihi0 = T_LEN - 1;
    int ilo1 = (int)ceil(((double)j1 / 63.0 - bw) * 511.0); if (ilo1 < 0) ilo1 = 0;
    int ihi1 = (int)floor(((double)j1 / 63.0 + bw) * 511.0); if (ihi1 > T_LEN - 1) ihi1 = T_LEN - 1;
    const unsigned span0 = (unsigned)(ihi0 - ilo0);
    const unsigned span1 = (unsigned)(ihi1 - ilo1);
    int ii0 = -j0 - ilo0;              // (i - ilo), incremented every step
    int ii1 = -j1 - ilo1;

    // Incremental LDS ring cursors: slot(d) = (d - j) mod RING, so each
    // pointer advances by K_LEN floats per step with a single wrap test.
    // While i = d-j is in [0, T) it points at row i's slot; otherwise at a
    // valid-but-stale slot whose value is masked to BIG anyway.
    float* p0 = &ldsC[((RING - j0) % RING) * K_LEN + j0];
    float* p1 = &ldsC[((RING - j1) % RING) * K_LEN + j1];
    float* const pend = &ldsC[RINGSZ];

    float d1_0 = BIGV, d1_1 = BIGV;    // diagonal d-1 values for j0, j1
    float d2_0 = BIGV;                 // diagonal d-2 value for j0
    float nb_prev = BIGV;              // shfl_up(d1_1) from the previous step
                                       //   == shfl_up(d2_1) for this step

    int sbase = 0;                     // uniform: (strip first row) mod RING

    for (int dbase = 0; dbase < N_DIAG; dbase += 16) {
        // ---- Produce cost strip rows [dbase, dbase+16) via 4 WMMA tiles
        // (uniform branch; EXEC all ones as WMMA requires). ----
        if (dbase < T_LEN) {
            float xv = x[b * T_LEN + dbase + lc];
            // A layout (16x4 f32, ISA 7.12.2): lanes 0-15: VGPR0=K0, VGPR1=K1;
            // lanes 16-31: VGPR0=K2, VGPR1=K3.  A[m,:] = [x^2, x, 1, 0]
            v2f a;
            a.x = (lane < 16) ? xv * xv : 1.0f;
            a.y = (lane < 16) ? xv : 0.0f;
            const int rbase = (lane < 16) ? 0 : 8;
            // sbase in {0,16,32,48,64} and rbase+r <= 15  ->  no ring wrap.
            const int basei = (sbase + rbase) * K_LEN + lc;
#pragma unroll
            for (int t = 0; t < 4; ++t) {
                v8f acc = {};
                acc = __builtin_amdgcn_wmma_f32_16x16x4_f32(
                    false, a, false, bvec[t], (short)0, acc, false, false);
                // C/D layout: lane n (0-15): VGPR r -> (M=r,  N=n)
                //             lane n+16    : VGPR r -> (M=8+r,N=n)
#pragma unroll
                for (int r = 0; r < 8; ++r) {
                    ldsC[basei + t * 16 + r * K_LEN] = acc[r];
                }
            }
            sbase += 16; if (sbase == RING) sbase = 0;
        }

        const int dend = (dbase + 16 < N_DIAG) ? dbase + 16 : N_DIAG;
        for (int d = dbase; d < dend; ++d) {
            // ---- Anti-diagonal DP step d: cells (i, j) with i + j == d. ----
            // Lane l hits banks 2l / 2l+1 -> conflict-free.
            float c0 = *p0;
            float c1 = *p1;

            // Single shuffle per step: shfl_up(d2_1) == last step's result.
            float nb_d1 = __shfl_up(d1_1, 1, 32);

            // Even column j0: up = D[i-1,j0] (=d1_0), left/diag from neighbor.
            float left0 = (lane == 0) ? BIGV : nb_d1;
            float diag0 = (lane == 0) ? ((d == 0) ? 0.0f : BIGV) : nb_prev;  // D[-1,-1]=0
            float n0 = c0 + softmin3_b2(d1_0, left0, diag0);
            n0 = ((unsigned)ii0 <= span0) ? n0 : BIGV;

            // Odd column j1: left = D[i,j0] (=d1_0), diag = D[i-1,j0] (=d2_0).
            float n1 = c1 + softmin3_b2(d1_1, d1_0, d2_0);
            n1 = ((unsigned)ii1 <= span1) ? n1 : BIGV;

            d2_0 = d1_0; d1_0 = n0; d1_1 = n1; nb_prev = nb_d1;
            ++ii0; ++ii1;
            p0 += K_LEN; if (p0 >= pend) p0 -= RINGSZ;
            p1 += K_LEN; if (p1 >= pend) p1 -= RINGSZ;
        }
    }

    // Final cell D[T-1, K-1] lives in lane 31's odd column after step 574.
    // Convert base-2 accumulation back to nats (* ln2) and average by T.
    if (lane == 31) {
        out[bf] = d1_1 * (LN2 / (float)T_LEN);
    }
}

extern "C" void kernel_launch(void* const* d_in, const int* in_sizes, int n_in,
                              void* d_out, int out_size, void* d_ws, size_t ws_size,
                              hipStream_t stream) {
    (void)in_sizes; (void)n_in; (void)d_ws; (void)ws_size; (void)out_size;
    const float* x      = (const float*)d_in[0];
    const float* protos = (const float*)d_in[1];
    float*       out    = (float*)d_out;

    dim3 grid(B_NUM * F_NUM);   // one wave32 workgroup per (b, f) problem
    dim3 block(32);
    dtw_softdtw_kernel<<<grid, block, 0, stream>>>(x, protos, out);
}